// Quantizer_20753281974677
// MI455X (gfx1250) — compile-verified
//
#include <hip/hip_runtime.h>

typedef __attribute__((ext_vector_type(2))) float v2f;
typedef __attribute__((ext_vector_type(4))) float v4f;
typedef __attribute__((ext_vector_type(8))) float v8f;

#define B_ 4
#define H_ 16
#define L_ 4096
#define D_ 64
#define S_ 128
#define CPAD 68  // padded LDS row stride (floats): bank step 4/row -> conflict-free

// One block = 8 waves = 256 threads, handles 128 tokens of one (b,h).
// Wave w: rows [blockIdx.x*128 + 16w, +16). Scores via V_WMMA_F32_16X16X4_F32.
__global__ __launch_bounds__(256) void quant_argmax_onehot(
    const float* __restrict__ x, const float* __restrict__ c,
    float* __restrict__ out)
{
    __shared__ float sC[S_ * CPAD];  // 34.8 KB of the 320 KB WGP LDS

    const int tid = threadIdx.x;
    const int bh  = blockIdx.y;          // b*H + h
    const int h   = bh & (H_ - 1);

    // ---- stage codebook c[h] (128x64 f32 = 32KB) into padded LDS ----
    {
        const float* src = c + (size_t)h * S_ * D_;
        // 128 rows * 16 float4 per row = 2048 float4 chunks, 8 per thread
        #pragma unroll
        for (int j = 0; j < 8; ++j) {
            int q   = tid + j * 256;
            int row = q >> 4;            // 0..127
            int w4  = q & 15;            // float4 within row
            v4f v = *reinterpret_cast<const v4f*>(src + row * D_ + w4 * 4);
            *reinterpret_cast<v4f*>(sC + row * CPAD + w4 * 4) = v;
        }
    }
    __syncthreads();

    const int wave  = tid >> 5;
    const int lane  = tid & 31;
    const int lhalf = lane >> 4;   // 0: lanes 0-15, 1: lanes 16-31
    const int l15   = lane & 15;

    const int rowbase = blockIdx.x * 128 + wave * 16;

    // ---- load A tile per ISA 16x4 f32 layout: lane holds row l15,
    //      (v0,v1) = K = 4k + 2*lhalf + {0,1}  -> one 8B load per K-step ----
    const float* xrow = x + ((size_t)bh * L_ + rowbase + l15) * D_ + lhalf * 2;
    v2f a[16];
    #pragma unroll
    for (int k = 0; k < 16; ++k)
        a[k] = *reinterpret_cast<const v2f*>(xrow + k * 4);

    float bestv[8];
    int   besti[8];
    #pragma unroll
    for (int i = 0; i < 8; ++i) { bestv[i] = -3.4e38f; besti[i] = 0; }

    // ---- 8 N-tiles of 16 codewords; K=64 in 16 steps of 4 ----
    #pragma unroll
    for (int nt = 0; nt < 8; ++nt) {
        const float* bbase = sC + (nt * 16 + l15) * CPAD + lhalf * 2;
        v8f acc = {};
        #pragma unroll
        for (int k = 0; k < 16; ++k) {
            v2f bf = *reinterpret_cast<const v2f*>(bbase + k * 4);
            acc = __builtin_amdgcn_wmma_f32_16x16x4_f32(
                false, a[k], false, bf, (short)0, acc, false, false);
        }
        const int scol = nt * 16 + l15;  // this lane's codeword column
        #pragma unroll
        for (int i = 0; i < 8; ++i) {
            // strict > keeps lowest index on ties (s increases with nt)
            if (acc[i] > bestv[i]) { bestv[i] = acc[i]; besti[i] = scol; }
        }
    }

    // ---- argmax reduction across each 16-lane half (masks stay in-half) ----
    #pragma unroll
    for (int m = 1; m <= 8; m <<= 1) {
        #pragma unroll
        for (int i = 0; i < 8; ++i) {
            float ov = __shfl_xor(bestv[i], m, 32);
            int   oi = __shfl_xor(besti[i], m, 32);
            if (ov > bestv[i] || (ov == bestv[i] && oi < besti[i])) {
                bestv[i] = ov; besti[i] = oi;
            }
        }
    }

    // ---- one-hot writes: 512B contiguous per row, NT float4 stores ----
    const size_t outbase = ((size_t)bh * L_ + rowbase) * S_;
    const int col = lane * 4;
    #pragma unroll
    for (int i = 0; i < 8; ++i) {
        int ilo = __shfl(besti[i], 0, 32);   // row rowbase + i     (lanes 0-15)
        int ihi = __shfl(besti[i], 16, 32);  // row rowbase + i + 8 (lanes 16-31)
        v4f vlo = { col == ilo ? 1.0f : 0.0f, col + 1 == ilo ? 1.0f : 0.0f,
                    col + 2 == ilo ? 1.0f : 0.0f, col + 3 == ilo ? 1.0f : 0.0f };
        v4f vhi = { col == ihi ? 1.0f : 0.0f, col + 1 == ihi ? 1.0f : 0.0f,
                    col + 2 == ihi ? 1.0f : 0.0f, col + 3 == ihi ? 1.0f : 0.0f };
        __builtin_nontemporal_store(
            vlo, reinterpret_cast<v4f*>(out + outbase + (size_t)i * S_) + lane);
        __builtin_nontemporal_store(
            vhi, reinterpret_cast<v4f*>(out + outbase + (size_t)(i + 8) * S_) + lane);
    }
}

// Second tuple output: pass c through to the tail of d_out.
__global__ __launch_bounds__(256) void copy_codebook(
    const float* __restrict__ c, float* __restrict__ out_tail)
{
    size_t i = (size_t)blockIdx.x * 256 + threadIdx.x;  // float4 index
    v4f v = reinterpret_cast<const v4f*>(c)[i];
    __builtin_nontemporal_store(v, reinterpret_cast<v4f*>(out_tail) + i);
}

extern "C" void kernel_launch(void* const* d_in, const int* in_sizes, int n_in,
                              void* d_out, int out_size, void* d_ws, size_t ws_size,
                              hipStream_t stream)
{
    const float* x = (const float*)d_in[0];   // [B,H,L,D] f32
    const float* c = (const float*)d_in[1];   // [H,S,D]  f32
    float* out = (float*)d_out;               // onehot [B,H,L,S] ++ c [H,S,D]

    dim3 grid(L_ / 128, B_ * H_);             // 32 x 64 blocks
    quant_argmax_onehot<<<grid, 256, 0, stream>>>(x, c, out);

    const int c4 = (H_ * S_ * D_) / 4;        // 32768 float4s
    copy_codebook<<<c4 / 256, 256, 0, stream>>>(
        c, out + (size_t)B_ * H_ * L_ * S_);
}